// CSRN_79164837200606
// MI455X (gfx1250) — compile-verified
//
#include <hip/hip_runtime.h>
#include <math.h>

typedef __attribute__((ext_vector_type(2))) float v2f;
typedef __attribute__((ext_vector_type(8))) float v8f;

#define B_ 16
#define C_ 64
#define H_ 256
#define W_ 256
#define G_ 3

// Fast sigmoid/tanh: v_exp + v_rcp (v_rcp_f32 is ~1ulp; rcp(inf)=0 so both
// saturate exactly to {0,1} / {-1,1} with no NaN paths).
__device__ __forceinline__ float sigm_fast(float v) {
    return __builtin_amdgcn_rcpf(1.0f + __expf(-v));
}
__device__ __forceinline__ float tanh_fast(float v) {
    return 1.0f - 2.0f * __builtin_amdgcn_rcpf(__expf(2.0f * v) + 1.0f);
}

// ---------------------------------------------------------------------------
// Kernel 0: repack conv weights channel-major so the conv kernel can read them
// through a block-uniform pointer (-> scalar s_load, SGPR operands in FMAs).
// wpack[c*36 + g*3 + k]      = conv_rows_w[g][c][k]
// wpack[c*36 + 18 + g*3 + k] = conv_cols_w[g][c][k]
// ---------------------------------------------------------------------------
__global__ __launch_bounds__(256) void repack_w_kernel(
    const float* __restrict__ wr, const float* __restrict__ wc,
    float* __restrict__ wpack)
{
    const int i = blockIdx.x * 256 + threadIdx.x;
    if (i >= C_ * 36) return;
    const int c = i / 36, r = i % 36;
    const int half = r / 18;
    const int g = (r % 18) / 3;
    const int k = r % 3;
    const float* src = half ? wc : wr;
    wpack[i] = src[(g * C_ + c) * 3 + k];
}

// ---------------------------------------------------------------------------
// Kernel 1: fused row-conv + col-conv (+tanh), one pass over x.
// 32x32 output tile per block, 2x2 micro-tile per thread, 34x34 halo in LDS.
// Weights come from wpack via uniform scalar loads (no LDS weight traffic).
//   crow[((b*H+h)*W+w)*3+g]   (row-conv over width)
//   ccol[((b*W+w)*H+h)*3+g]   (col-conv over height, transposed for GRU reads)
// ---------------------------------------------------------------------------
#define TS 32
#define HS 34   // tile + halo

__global__ __launch_bounds__(256) void conv_both_kernel(
    const float* __restrict__ x,
    const float* __restrict__ wpack,
    const float* __restrict__ br, const float* __restrict__ bc,
    float* __restrict__ crow, float* __restrict__ ccol)
{
    __shared__ float tile[HS * HS];

    const int tid = threadIdx.x;
    const int bidx = blockIdx.x;                 // (W/32)*(H/32)*B tiles
    const int twx = bidx % (W_ / TS);
    const int thx = (bidx / (W_ / TS)) % (H_ / TS);
    const int b   = bidx / ((W_ / TS) * (H_ / TS));
    const int h0 = thx * TS, w0 = twx * TS;
    const int qy = tid >> 4, qx = tid & 15;      // 16x16 quads of 2x2

    float aR[2][2][G_], aC[2][2][G_];
#pragma unroll
    for (int dy = 0; dy < 2; ++dy)
#pragma unroll
        for (int dx = 0; dx < 2; ++dx)
#pragma unroll
            for (int g = 0; g < G_; ++g) { aR[dy][dx][g] = 0.f; aC[dy][dx][g] = 0.f; }

    const float* xb = x + (size_t)b * C_ * H_ * W_;

    for (int c = 0; c < C_; ++c) {
        __syncthreads();
        const float* xch = xb + (size_t)c * H_ * W_;
        for (int i = tid; i < HS * HS; i += 256) {
            const int r  = i / HS;
            const int cc = i % HS;
            const int gh = h0 - 1 + r;
            const int gw = w0 - 1 + cc;
            float v = 0.f;
            if ((unsigned)gh < (unsigned)H_ && (unsigned)gw < (unsigned)W_)
                v = xch[gh * W_ + gw];
            tile[i] = v;
        }
        // prefetch next channel's plane near this thread's region
        if (c + 1 < C_) {
            const int gh = h0 - 1 + (tid / HS);
            const int gw = w0 - 1 + (tid % HS);
            if ((unsigned)gh < (unsigned)H_ && (unsigned)gw < (unsigned)W_)
                __builtin_prefetch(xch + H_ * W_ + gh * W_ + gw, 0, 3);
        }
        __syncthreads();

        // 12 unique LDS reads feed 4 outputs x (row+col) convs
        const float* A  = &tile[(2 * qy + 1) * HS + 2 * qx];  // center row dy=0
        const float a0 = A[0], a1 = A[1], a2 = A[2], a3 = A[3];
        const float* Bv = A + HS;                             // center row dy=1
        const float b0 = Bv[0], b1 = Bv[1], b2 = Bv[2], b3 = Bv[3];
        const float* Tp = &tile[(2 * qy) * HS + 2 * qx];      // top halo row
        const float t1 = Tp[1], t2 = Tp[2];
        const float* Bt = &tile[(2 * qy + 3) * HS + 2 * qx];  // bottom halo row
        const float u1 = Bt[1], u2 = Bt[2];

        const float* wp = wpack + c * 36;     // uniform -> scalar loads
#pragma unroll
        for (int g = 0; g < G_; ++g) {
            const float r0 = wp[g * 3 + 0], r1 = wp[g * 3 + 1], r2 = wp[g * 3 + 2];
            aR[0][0][g] += r0 * a0 + r1 * a1 + r2 * a2;
            aR[0][1][g] += r0 * a1 + r1 * a2 + r2 * a3;
            aR[1][0][g] += r0 * b0 + r1 * b1 + r2 * b2;
            aR[1][1][g] += r0 * b1 + r1 * b2 + r2 * b3;
            const float c0 = wp[18 + g * 3 + 0], c1 = wp[18 + g * 3 + 1], c2 = wp[18 + g * 3 + 2];
            aC[0][0][g] += c0 * t1 + c1 * a1 + c2 * b1;
            aC[0][1][g] += c0 * t2 + c1 * a2 + c2 * b2;
            aC[1][0][g] += c0 * a1 + c1 * b1 + c2 * u1;
            aC[1][1][g] += c0 * a2 + c1 * b2 + c2 * u2;
        }
    }

    const float rb0 = br[0], rb1 = br[1], rb2 = br[2];
    const float cb0 = bc[0], cb1 = bc[1], cb2 = bc[2];
#pragma unroll
    for (int dy = 0; dy < 2; ++dy) {
#pragma unroll
        for (int dx = 0; dx < 2; ++dx) {
            const int oh = h0 + 2 * qy + dy;
            const int ow = w0 + 2 * qx + dx;
            const size_t pr = ((size_t)b * H_ + oh) * W_ + ow;
            crow[pr * 3 + 0] = tanh_fast(aR[dy][dx][0] + rb0);
            crow[pr * 3 + 1] = tanh_fast(aR[dy][dx][1] + rb1);
            crow[pr * 3 + 2] = tanh_fast(aR[dy][dx][2] + rb2);
            const size_t pc = ((size_t)b * W_ + ow) * H_ + oh;
            ccol[pc * 3 + 0] = tanh_fast(aC[dy][dx][0] + cb0);
            ccol[pc * 3 + 1] = tanh_fast(aC[dy][dx][1] + cb1);
            ccol[pc * 3 + 2] = tanh_fast(aC[dy][dx][2] + cb2);
        }
    }
}

// ---------------------------------------------------------------------------
// Kernel 2: both GRU scans (hidden=3, input=3). Blocks [0,half) row sweep
// (scan over H, batch B*W); blocks [half,2*half) col sweep (scan over W,
// batch B*H). 4-deep explicit load pipeline hides per-step load latency.
// ---------------------------------------------------------------------------
__global__ __launch_bounds__(256) void gru_kernel(
    const float* __restrict__ crow, const float* __restrict__ ccol,
    const float* __restrict__ wih_r, const float* __restrict__ whh_r,
    const float* __restrict__ bih_r, const float* __restrict__ bhh_r,
    const float* __restrict__ wih_c, const float* __restrict__ whh_c,
    const float* __restrict__ bih_c, const float* __restrict__ bhh_c,
    float* __restrict__ rowd, float* __restrict__ cold)
{
    const int half = gridDim.x >> 1;
    const bool isRow = (blockIdx.x < (unsigned)half);
    const int idx = (isRow ? blockIdx.x : (blockIdx.x - half)) * 256 + threadIdx.x;

    const float *in, *wih, *whh, *bih, *bhh;
    float* out;
    size_t ibase, istep, obase, ostep;
    int T;

    if (isRow) {
        const int b = idx / W_, w = idx % W_;
        in = crow; out = rowd;
        wih = wih_r; whh = whh_r; bih = bih_r; bhh = bhh_r;
        ibase = ((size_t)b * H_ * W_ + w) * 3;     // crow[(b*H+h)*W+w]
        istep = (size_t)W_ * 3;
        obase = ibase; ostep = istep;              // rowd same layout
        T = H_;
    } else {
        const int b = idx / H_, h = idx % H_;
        in = ccol; out = cold;
        wih = wih_c; whh = whh_c; bih = bih_c; bhh = bhh_c;
        ibase = ((size_t)b * W_ * H_ + h) * 3;     // ccol[(b*W+w)*H+h]
        istep = (size_t)H_ * 3;
        obase = ((size_t)b * H_ + h) * (size_t)W_ * 3;  // cold[((b*H+h)*W+w)*3]
        ostep = 3;
        T = W_;
    }

    float Wi[9][3], Wh[9][3], Bi[9], Bh[9];
#pragma unroll
    for (int j = 0; j < 9; ++j) {
        Bi[j] = bih[j];
        Bh[j] = bhh[j];
#pragma unroll
        for (int i = 0; i < 3; ++i) {
            Wi[j][i] = wih[j * 3 + i];
            Wh[j][i] = whh[j * 3 + i];
        }
    }

    const float* pin = in + ibase;
    float a0 = pin[0], a1 = pin[1], a2 = pin[2];
    float b0 = pin[istep + 0], b1 = pin[istep + 1], b2 = pin[istep + 2];
    float c0 = pin[2 * istep + 0], c1 = pin[2 * istep + 1], c2 = pin[2 * istep + 2];
    float d0 = pin[3 * istep + 0], d1 = pin[3 * istep + 1], d2 = pin[3 * istep + 2];

    float hs0 = 0.f, hs1 = 0.f, hs2 = 0.f;

    for (int t = 0; t < T; ++t) {
        const float X0 = a0, X1 = a1, X2 = a2;
        a0 = b0; a1 = b1; a2 = b2;
        b0 = c0; b1 = c1; b2 = c2;
        c0 = d0; c1 = d1; c2 = d2;
        const int nt = t + 4;
        if (nt < T) {
            const float* pn = pin + (size_t)nt * istep;
            d0 = pn[0]; d1 = pn[1]; d2 = pn[2];
        } else {
            d0 = 0.f; d1 = 0.f; d2 = 0.f;
        }

        float gi[9], gh[9];
#pragma unroll
        for (int j = 0; j < 9; ++j)
            gi[j] = Bi[j] + X0 * Wi[j][0] + X1 * Wi[j][1] + X2 * Wi[j][2];
#pragma unroll
        for (int j = 0; j < 9; ++j)
            gh[j] = Bh[j] + hs0 * Wh[j][0] + hs1 * Wh[j][1] + hs2 * Wh[j][2];

        const float r0 = sigm_fast(gi[0] + gh[0]);
        const float r1 = sigm_fast(gi[1] + gh[1]);
        const float r2 = sigm_fast(gi[2] + gh[2]);
        const float z0 = sigm_fast(gi[3] + gh[3]);
        const float z1 = sigm_fast(gi[4] + gh[4]);
        const float z2 = sigm_fast(gi[5] + gh[5]);
        const float n0 = tanh_fast(gi[6] + r0 * gh[6]);
        const float n1 = tanh_fast(gi[7] + r1 * gh[7]);
        const float n2 = tanh_fast(gi[8] + r2 * gh[8]);
        hs0 = (1.f - z0) * n0 + z0 * hs0;
        hs1 = (1.f - z1) * n1 + z1 * hs1;
        hs2 = (1.f - z2) * n2 + z2 * hs2;

        float* po = out + obase + (size_t)t * ostep;
        po[0] = hs0; po[1] = hs1; po[2] = hs2;
    }
}

// ---------------------------------------------------------------------------
// Kernel 3: combine with V_WMMA_F32_16X16X4_F32 (fp32 WMMA, K=6 padded to 8).
// D(M=16 channels x N=16 positions) = Wc_tile x g^T + bias; 4 channel tiles,
// 2 K-steps each -> 8 WMMAs per wave of 16 positions. Lane = w-position ->
// coalesced 64B half-wave stores of the (B,C,H,W) output.
// ---------------------------------------------------------------------------
__global__ __launch_bounds__(256) void combine_kernel(
    const float* __restrict__ rowd, const float* __restrict__ cold,
    const float* __restrict__ Wcw, const float* __restrict__ bcb,
    float* __restrict__ out)
{
    const int lane = threadIdx.x & 31;
    const int wv   = threadIdx.x >> 5;
    const int gw   = blockIdx.x * 8 + wv;
    const int pbase = gw * 16;

    const int b   = pbase / (H_ * W_);
    const int rem = pbase % (H_ * W_);
    const int h   = rem / W_;
    const int w0  = rem % W_;

    const int nidx  = lane & 15;               // N = position within group
    const int khalf = lane >> 4;               // lanes 16-31 hold upper K pair
    const int k0    = khalf * 2;
    const size_t p  = (size_t)pbase + nidx;

    auto gval = [&](int k) -> float {
        if (k < 3) return rowd[p * 3 + k];
        if (k < 6) return cold[p * 3 + (k - 3)];
        return 0.f;
    };
    v2f Bm0 = { gval(k0),     gval(k0 + 1) };      // K = 0..3
    v2f Bm1 = { gval(k0 + 4), gval(k0 + 5) };      // K = 4..7 (6,7 -> 0)

    const int m    = lane & 15;                // M = channel within tile
    const int mofs = khalf * 8;                // C/D: lanes>=16 hold M=r+8

    auto wget = [&](int c, int k) -> float {
        return (k < 6) ? Wcw[c * 6 + k] : 0.f;
    };

#pragma unroll
    for (int t = 0; t < 4; ++t) {
        const int cbase = t * 16;
        const int cA = cbase + m;
        v2f Am0 = { wget(cA, k0),     wget(cA, k0 + 1) };
        v2f Am1 = { wget(cA, k0 + 4), wget(cA, k0 + 5) };

        v8f acc;
#pragma unroll
        for (int r = 0; r < 8; ++r)
            acc[r] = bcb[cbase + r + mofs];

        acc = __builtin_amdgcn_wmma_f32_16x16x4_f32(
            false, Am0, false, Bm0, (short)0, acc, false, false);
        acc = __builtin_amdgcn_wmma_f32_16x16x4_f32(
            false, Am1, false, Bm1, (short)0, acc, false, false);

#pragma unroll
        for (int r = 0; r < 8; ++r) {
            const float v = sigm_fast(acc[r]);
            const int c_r = cbase + r + mofs;
            const size_t off = (((size_t)b * C_ + c_r) * H_ + h) * W_ + w0 + nidx;
            out[off] = v;
        }
    }
}

// ---------------------------------------------------------------------------
extern "C" void kernel_launch(void* const* d_in, const int* in_sizes, int n_in,
                              void* d_out, int out_size, void* d_ws, size_t ws_size,
                              hipStream_t stream) {
    const float* x     = (const float*)d_in[0];
    const float* crw   = (const float*)d_in[1];
    const float* crb   = (const float*)d_in[2];
    const float* ccw   = (const float*)d_in[3];
    const float* ccb   = (const float*)d_in[4];
    const float* grwih = (const float*)d_in[5];
    const float* grwhh = (const float*)d_in[6];
    const float* grbih = (const float*)d_in[7];
    const float* grbhh = (const float*)d_in[8];
    const float* gcwih = (const float*)d_in[9];
    const float* gcwhh = (const float*)d_in[10];
    const float* gcbih = (const float*)d_in[11];
    const float* gcbhh = (const float*)d_in[12];
    const float* cmbw  = (const float*)d_in[13];
    const float* cmbb  = (const float*)d_in[14];
    float* out = (float*)d_out;

    const size_t NP = (size_t)B_ * H_ * W_;    // 1,048,576 positions
    float* ws    = (float*)d_ws;
    float* crow  = ws;                         // (B,H,W,G)
    float* ccol  = ws + 1 * NP * 3;            // (B,W,H,G)
    float* rowd  = ws + 2 * NP * 3;            // (B,H,W,G)
    float* cold  = ws + 3 * NP * 3;            // (B,H,W,G)
    float* wpack = ws + 4 * NP * 3;            // (C, 36)

    // 0) repack conv weights channel-major (C_*36 = 2304 elems)
    repack_w_kernel<<<dim3((C_ * 36 + 255) / 256), dim3(256), 0, stream>>>(
        crw, ccw, wpack);

    // 1) fused convs: 32x32 tiles -> B*(H/32)*(W/32) = 1024 blocks
    conv_both_kernel<<<dim3(B_ * (H_ / TS) * (W_ / TS)), dim3(256), 0, stream>>>(
        x, wpack, crb, ccb, crow, ccol);

    // 2) both GRU sweeps in one launch: 16 blocks rows + 16 blocks cols
    gru_kernel<<<dim3(32), dim3(256), 0, stream>>>(
        crow, ccol, grwih, grwhh, grbih, grbhh,
        gcwih, gcwhh, gcbih, gcbhh, rowd, cold);

    // 3) WMMA combine + sigmoid: NP/(8 waves * 16 pos) = 8192 blocks
    combine_kernel<<<dim3((int)(NP / 128)), dim3(256), 0, stream>>>(
        rowd, cold, cmbw, cmbb, out);
}